// PSLoss_83511344103707
// MI455X (gfx1250) — compile-verified
//
#include <hip/hip_runtime.h>
#include <hip/hip_bf16.h>

typedef float v2f __attribute__((ext_vector_type(2)));
typedef float v8f __attribute__((ext_vector_type(8)));

#define B_    64
#define T_    336
#define F_    321
#define NSER  (B_ * F_)            // 20544 series
#define SPB   16                   // series per block (WMMA M dim)
#define NBLK  (NSER / SPB)         // 1284 (exact)
#define NFREQ 168                  // rfft bins 1..168 (DC dropped)
#define NTILE 11                   // ceil(176/16) freq tiles of 16
#define KD    84                   // 336 / 4 WMMA K-steps
#define TGS   338                  // LDS row stride for raw series
#define CSS   337                  // prefix-sum length (T+1)
#define CTAB  1008                 // extended cos table: no mod for +kf / +252 offsets
#define EPSL  1e-8f

__global__ __launch_bounds__(128) void psloss_main(const float* __restrict__ fc,
                                                   const float* __restrict__ tg,
                                                   float* __restrict__ ws) {
    __shared__ float tgL[SPB][TGS];
    __shared__ float fcL[SPB][TGS];
    __shared__ float cs[5][SPB][CSS];        // prefix sums: t, f, tt, ff, tf
    __shared__ float ampL[SPB][NFREQ];
    __shared__ float costbl[CTAB];
    __shared__ float red[128];
    __shared__ float ptot[SPB];
    __shared__ int   pcnt[SPB];

    const int tid = threadIdx.x;
    const int blk = blockIdx.x;

    // extended twiddle table: cos(2*pi*(i mod 336)/336) for i in [0,1008)
    for (int i = tid; i < CTAB; i += 128) {
        int ii = i;
        if (ii >= 672) ii -= 672; else if (ii >= T_) ii -= T_;
        costbl[i] = __cosf(6.28318530717958647692f * (float)ii / (float)T_);
    }

    // ---- Phase 1: load 16 series into LDS (coalesced over feature dim) + MSE partial
    float msep = 0.0f;
    for (int e = tid; e < SPB * T_; e += 128) {
        int m = e & 15;
        int t = e >> 4;
        int s = blk * SPB + m;
        int b = s / F_;
        int f = s - b * F_;
        int g = (b * T_ + t) * F_ + f;
        float tv = tg[g];
        float fv = fc[g];
        tgL[m][t] = tv;
        fcL[m][t] = fv;
        float d = fv - tv;
        msep += d * d;
    }
    __syncthreads();

    // ---- Phase 2 (concurrent): waves 0-2 do the WMMA DFT, wave 3 does prefix sums.
    // DFT reads tgL/costbl, writes ampL; prefix reads tgL/fcL, writes cs -> disjoint.
    const int lane = tid & 31;
    const int wave = tid >> 5;

    if (wave < 3) {
        // DFT amplitudes via V_WMMA_F32_16X16X4_F32:
        // D[16 series x 16 freqs] += A[16x4 time] * B[4x16]; 84 K-steps; 11 tiles over 3 waves
        const int mrow = lane & 15;             // A: M (series) / B: N (freq) per ISA layout
        const int koff = (lane >> 4) << 1;      // lanes 0-15: K=0,1 ; lanes 16-31: K=2,3
        const int half = (lane >> 4) << 3;      // C/D rows: lanes 0-15 -> M=r, 16-31 -> M=8+r

        for (int ft = wave; ft < NTILE; ft += 3) {
            int kf = ft * 16 + mrow + 1;        // frequency bin (1-based), <= 176
            v8f accre = {};
            v8f accim = {};
            int cur  = (kf * koff) % T_;        // (kf * t) mod 336, incremental
            int step = (kf * 4) % T_;
            for (int kk = 0; kk < KD; ++kk) {
                int tb = kk * 4;
                v2f a;
                a.x = tgL[mrow][tb + koff];
                a.y = tgL[mrow][tb + koff + 1];
                // extended table: only 'cur' wraps; offsets stay in-range (<1008)
                v2f bc, bs;
                bc.x = costbl[cur];
                bc.y = costbl[cur + kf];
                bs.x = costbl[cur + 252];
                bs.y = costbl[cur + 252 + kf];
                accre = __builtin_amdgcn_wmma_f32_16x16x4_f32(false, a, false, bc,
                                                              (short)0, accre, false, false);
                accim = __builtin_amdgcn_wmma_f32_16x16x4_f32(false, a, false, bs,
                                                              (short)0, accim, false, false);
                cur += step; if (cur >= T_) cur -= T_;
            }
            int n = ft * 16 + mrow;             // 0-based bin index (bin k = n+1)
            if (n < NFREQ) {
#pragma unroll
                for (int r = 0; r < 8; ++r) {
                    float re = accre[r];
                    float im = accim[r];
                    ampL[half + r][n] = re * re + im * im;  // squared amp: same ranking
                }
            }
        }
    } else {
        // Five prefix-sum arrays per series: 80 (series,stat) pairs over 32 lanes
        for (int pid = lane; pid < 5 * SPB; pid += 32) {
            int m = pid & 15;
            int w = pid >> 4;                   // 0:t 1:f 2:tt 3:ff 4:tf
            float acc = 0.0f;
            cs[w][m][0] = 0.0f;
            for (int t = 0; t < T_; ++t) {
                float tv = tgL[m][t];
                float fv = fcL[m][t];
                float x = (w == 0) ? tv
                        : (w == 1) ? fv
                        : (w == 2) ? tv * tv
                        : (w == 3) ? fv * fv
                        :            tv * fv;
                acc += x;
                cs[w][m][t + 1] = acc;
            }
        }
    }
    __syncthreads();

    // ---- Phase 3: top-3 bins -> periods -> patches -> losses (one lane per series)
    float ltot = 0.0f;
    int   lcnt = 0;
    if (tid < SPB) {
        const int m = tid;
        auto patch = [&](int s0, int e0) {
            float fn  = (float)(e0 - s0);
            float St  = cs[0][m][e0] - cs[0][m][s0];
            float Sf  = cs[1][m][e0] - cs[1][m][s0];
            float Stt = cs[2][m][e0] - cs[2][m][s0];
            float Sff = cs[3][m][e0] - cs[3][m][s0];
            float Stf = cs[4][m][e0] - cs[4][m][s0];
            float mt = St / fn, mf = Sf / fn;
            float tvs = fmaxf(Stt - fn * mt * mt, 0.0f);
            float fvs = fmaxf(Sff - fn * mf * mf, 0.0f);
            float num = Stf - fn * mt * mf;
            float corr  = num / (sqrtf(tvs * fvs) + EPSL);
            float closs = 1.0f - fabsf(corr);
            float tvar  = tvs / (fn - 1.0f);
            float fvar  = fvs / (fn - 1.0f);
            float vloss = fabsf(tvar - fvar) / (tvar + EPSL);
            float mloss = fabsf(mt - mf) / (fabsf(mt) + EPSL);
            ltot += closs + vloss + mloss;      // W_CORR = W_VAR = W_MEAN = 1
            lcnt += 1;
        };

        // top-3 (stable: earlier index wins ties, matching lax.top_k)
        float v0 = -1.0f, v1 = -1.0f, v2 = -1.0f;
        int   r0 = 0, r1 = 0, r2 = 0;
        for (int n = 0; n < NFREQ; ++n) {
            float a = ampL[m][n];
            if (a > v0)      { v2 = v1; r2 = r1; v1 = v0; r1 = r0; v0 = a; r0 = n; }
            else if (a > v1) { v2 = v1; r2 = r1; v1 = a; r1 = n; }
            else if (a > v2) { v2 = a; r2 = n; }
        }
        int p[3];
        p[0] = T_ / (r0 + 1);
        p[1] = T_ / (r1 + 1);
        p[2] = T_ / (r2 + 1);
        bool valid[3];
        valid[0] = (p[0] >= 5);
        valid[1] = (p[1] >= 5) && (p[1] != p[0]);
        valid[2] = (p[2] >= 5) && (p[2] != p[0]) && (p[2] != p[1]);
        bool anyv = valid[0] | valid[1] | valid[2];

        int  tails[3];
        bool tval[3];
#pragma unroll
        for (int j = 0; j < 3; ++j) { tails[j] = -1; tval[j] = false; }

        for (int j = 0; j < 3; ++j) {
            if (!valid[j]) continue;
            int pj = p[j];
            int nseg = T_ / pj;
            for (int i = 0; i < nseg; ++i)
                patch(i * pj, i * pj + pj);     // distinct lengths across periods -> unique
            int rs = nseg * pj;
            if (T_ - rs >= 5) { tails[j] = rs; tval[j] = true; }
        }
        // tails are the only possible duplicates: vs a dividing-period last segment, or earlier tail
        for (int j = 0; j < 3; ++j) {
            if (!tval[j]) continue;
            bool dup = false;
            for (int k = 0; k < 3; ++k)
                if (valid[k] && (T_ % p[k] == 0) && (tails[j] == T_ - p[k])) dup = true;
            for (int k = 0; k < j; ++k)
                if (tval[k] && tails[k] == tails[j]) dup = true;
            if (!dup) patch(tails[j], T_);
        }
        if (!anyv) patch(0, T_);                // fallback whole-series patch

        ptot[m] = ltot;
        pcnt[m] = lcnt;
    }

    // ---- Reductions + atomics
    red[tid] = msep;
    __syncthreads();
    for (int off = 64; off >= 1; off >>= 1) {
        if (tid < off) red[tid] += red[tid + off];
        __syncthreads();
    }
    if (tid == 0) {
        float pt = 0.0f;
        int   pc = 0;
        for (int i = 0; i < SPB; ++i) { pt += ptot[i]; pc += pcnt[i]; }
        atomicAdd(&ws[0], red[0]);
        atomicAdd(&ws[1], pt);
        atomicAdd(reinterpret_cast<int*>(ws) + 2, pc);
    }
}

__global__ void psloss_final(const float* __restrict__ ws, float* __restrict__ out) {
    float mse = ws[0] / (float)(B_ * T_ * F_);
    int   pc  = reinterpret_cast<const int*>(ws)[2];
    float avg = (pc > 0) ? (ws[1] / (float)pc) : 0.0f;
    out[0] = 0.5f * mse + 0.5f * avg;
}

extern "C" void kernel_launch(void* const* d_in, const int* in_sizes, int n_in,
                              void* d_out, int out_size, void* d_ws, size_t ws_size,
                              hipStream_t stream) {
    const float* fc = (const float*)d_in[0];   // forecast [B,T,F] f32
    const float* tg = (const float*)d_in[1];   // target   [B,T,F] f32
    float* ws = (float*)d_ws;                  // [0]=mse_sum [1]=patch_total [2]=patch_count(int)
    hipMemsetAsync(d_ws, 0, 3 * sizeof(float), stream);
    psloss_main<<<NBLK, 128, 0, stream>>>(fc, tg, ws);
    psloss_final<<<1, 1, 0, stream>>>(ws, (float*)d_out);
}